// PVBMLYLIFH_44143673868846
// MI455X (gfx1250) — compile-verified
//
#include <hip/hip_runtime.h>
#include <hip/hip_bf16.h>

// Problem constants (fixed by the reference).
#define T_STEPS 5
#define BS      32            // batch per time step (160 / 5)
#define CCH     128
#define HH      48
#define WW      48
#define HW      (HH * WW)     // 2304
#define HW4     (HW / 4)      // 576 float4 per image plane
#define SLAB4   (BS * CCH * HW4)  // float4 elements per time slab = 2,359,296
#define BLOCKS_PER_B ((CCH * HW4) / 256)  // 288

typedef __attribute__((ext_vector_type(4))) float v4f;

// ---------------------------------------------------------------------------
// Kernel 1: DropBlock mask.  bm[n,h,w] = 1 - maxpool7x7_SAME(mask_rand < gamma)
// Stage 1 uses CDNA5 async global->LDS copy (ASYNCcnt path): the raw 48x48
// plane (9.2 KB) is DMA'd into LDS without a VGPR round trip, then the
// threshold is fused into the separable horizontal max pass.
// One block per image n (160 blocks); HW == 2304 == 9 * 256 exactly.
// ---------------------------------------------------------------------------
__global__ __launch_bounds__(256) void blockmask_kernel(
    const float* __restrict__ mask_rand, float* __restrict__ bm)
{
    __shared__ float mraw[HW];   // raw mask_rand plane (async-copied)
    __shared__ float hmax[HW];   // horizontal 7-max of thresholded predicate

    const int n = blockIdx.x;
    const float gamma = (float)(0.2 / 49.0);   // DROP_RATE / BLOCK_SIZE^2
    const float* m = mask_rand + (size_t)n * HW;

    // Stage 1: async copy global -> LDS (9 x b32 per lane; EXEC all ones).
    // LDS byte address for the async op is the low 32 bits of the generic
    // pointer (ISA 10.2).
#pragma unroll
    for (int k = 0; k < HW / 256; ++k) {
        const int i = threadIdx.x + k * 256;
        unsigned lds_off = (unsigned)(unsigned long long)&mraw[i];
        const float* gp = m + i;
        asm volatile("global_load_async_to_lds_b32 %0, %1, off"
                     :: "v"(lds_off), "v"(gp)
                     : "memory");
    }
    asm volatile("s_wait_asynccnt 0" ::: "memory");   // own wave's DMAs done
    __syncthreads();                                  // all waves' DMAs visible

    // Stage 2: threshold fused with horizontal max over [w-3, w+3] (SAME pad).
    for (int i = threadIdx.x; i < HW; i += 256) {
        const int h = i / WW, w = i % WW;
        const int w0 = (w - 3 < 0) ? 0 : w - 3;
        const int w1 = (w + 3 > WW - 1) ? WW - 1 : w + 3;
        float mx = 0.0f;
        for (int ww = w0; ww <= w1; ++ww)
            mx = fmaxf(mx, (mraw[h * WW + ww] < gamma) ? 1.0f : 0.0f);
        hmax[i] = mx;
    }
    __syncthreads();

    // Stage 3: vertical max over [h-3, h+3], emit bm = 1 - pooled.
    for (int i = threadIdx.x; i < HW; i += 256) {
        const int h = i / WW, w = i % WW;
        const int h0 = (h - 3 < 0) ? 0 : h - 3;
        const int h1 = (h + 3 > HH - 1) ? HH - 1 : h + 3;
        float mx = 0.0f;
        for (int hh = h0; hh <= h1; ++hh)
            mx = fmaxf(mx, hmax[hh * WW + w]);
        bm[(size_t)n * HW + i] = 1.0f - mx;
    }
}

// ---------------------------------------------------------------------------
// Kernel 2: LIF scan over T=5, vectorized float4 (b128 VMEM), fully unrolled.
//   u  = (u >= v ? 0 : 0.5*u) + x_t          (step() of the OLD u)
//   o_t = (u >= v ? 1 : 0) * bm_t            (step() of the NEW u)
// Streams (x, out) are non-temporal (touch-once, 2 x 188.7 MB); bm stays
// temporal (reused by the 288 blocks covering each b-slab) and is prefetched.
// b derives from blockIdx only -> the 5 vth loads are scalar (s_load).
// ---------------------------------------------------------------------------
__global__ __launch_bounds__(256) void lif_kernel(
    const float* __restrict__ x, const float* __restrict__ vth,
    const float* __restrict__ bm, float* __restrict__ out)
{
    const int tid = blockIdx.x * 256 + threadIdx.x;   // [0, SLAB4)
    const int hw4 = tid % HW4;
    const int b   = blockIdx.x / BLOCKS_PER_B;        // uniform per block

    // Per-(t,b) thresholds: uniform addresses -> SMEM scalar loads.
    float v[T_STEPS];
#pragma unroll
    for (int t = 0; t < T_STEPS; ++t)
        v[t] = vth[(t * BS + b) * 3];                 // LAYER-1 == 0

    const v4f* __restrict__ x4  = (const v4f*)x;
    const v4f* __restrict__ bm4 = (const v4f*)bm;
    v4f*       __restrict__ o4  = (v4f*)out;

    // Warm the reused mask cachelines (global_prefetch_b8).
    __builtin_prefetch(&bm4[(long)b * HW4 + hw4], 0, 3);

    v4f u = {0.0f, 0.0f, 0.0f, 0.0f};
#pragma unroll
    for (int t = 0; t < T_STEPS; ++t) {
        const long xi = (long)t * SLAB4 + tid;
        v4f xv  = __builtin_nontemporal_load(&x4[xi]);                 // b128 NT
        v4f bmv = bm4[(long)(t * BS + b) * HW4 + hw4];                 // b128 RT (hot)
        v4f o;
#pragma unroll
        for (int i = 0; i < 4; ++i) {
            float uu = u[i];
            uu = ((uu >= v[t]) ? 0.0f : 0.5f * uu) + xv[i];            // TAU = 0.5
            u[i] = uu;
            o[i] = (uu >= v[t]) ? bmv[i] : 0.0f;                       // spike * bm
        }
        __builtin_nontemporal_store(o, &o4[xi]);                       // b128 NT
    }
}

// ---------------------------------------------------------------------------
extern "C" void kernel_launch(void* const* d_in, const int* in_sizes, int n_in,
                              void* d_out, int out_size, void* d_ws, size_t ws_size,
                              hipStream_t stream)
{
    const float* x         = (const float*)d_in[0];   // (160,128,48,48)
    const float* vth       = (const float*)d_in[1];   // (160,3)
    const float* mask_rand = (const float*)d_in[2];   // (160,48,48)
    float*       out       = (float*)d_out;
    float*       bm        = (float*)d_ws;            // 160*48*48 floats = 1.41 MB

    blockmask_kernel<<<T_STEPS * BS, 256, 0, stream>>>(mask_rand, bm);
    lif_kernel<<<SLAB4 / 256, 256, 0, stream>>>(x, vth, bm, out);
}